// OBBNMSAndReturnAsFlatResult_23716809409210
// MI455X (gfx1250) — compile-verified
//
#include <hip/hip_runtime.h>
#include <hip/hip_bf16.h>

#define BB 8
#define NN 2048
#define CC 80
#define FSTRIDE 32
#define CONF_T 0.25f
#define IOU_T 0.05f
#define SIGMA 3.0f
#define MAX_PRED 300
#define EPSF 1e-7f
#define LN2F 0.6931471805599453f
#define LOG2EF 1.4426950408889634f

typedef float v2f __attribute__((ext_vector_type(2)));
typedef float v8f __attribute__((ext_vector_type(8)));

static __device__ __forceinline__ v8f wmma4(v2f a, v2f b, v8f c) {
  // D = A(16x4 f32) * B(4x16 f32) + C(16x16 f32)
  return __builtin_amdgcn_wmma_f32_16x16x4_f32(
      /*neg_a=*/false, a, /*neg_b=*/false, b,
      /*c_mod=*/(short)0, c, /*reuse_a=*/false, /*reuse_b=*/false);
}

// raw-hardware transcendentals: inputs here are far from denormal range,
// so skip ocml's denorm guards (v_log_f32 is log2, v_exp_f32 is exp2)
static __device__ __forceinline__ float fast_ln(float x)  { return LN2F * __builtin_amdgcn_logf(x); }
static __device__ __forceinline__ float fast_exp(float x) { return __builtin_amdgcn_exp2f(x * LOG2EF); }

__global__ void k_zero(float* __restrict__ out, int n) {
  int i = blockIdx.x * blockDim.x + threadIdx.x;
  if (i < n) out[i] = 0.f;
}

// Per box: conf/label from scores, and the 28-entry bilinear feature vectors
// F (row/i side) and G (col/j side) so that for each of the three pairwise
// quantities  q[i,j] = sum_k F[i][k]*G[j][k]  over a K-chunk:
//   feats  0..11 : sa*dy^2 + sb*dx^2      (10 real + 2 zero pad)
//   feats 12..19 : -sc*dx*dy              (8 real)
//   feats 20..27 : sa*sb - sc^2           (5 real + 3 zero pad)
// sqrt(max(det,0)) goes to a dense side array (LDS-staged in the pass).
__global__ void k_prep(const float* __restrict__ boxes, const float* __restrict__ scores,
                       float* __restrict__ F, float* __restrict__ G,
                       float* __restrict__ conf, int* __restrict__ label,
                       float* __restrict__ sdv) {
  int i = blockIdx.x * blockDim.x + threadIdx.x;
  if (i >= BB * NN) return;

  const float* s = scores + (size_t)i * CC;
  float best = s[0]; int bi = 0;
  for (int c = 1; c < CC; ++c) { float v = s[c]; if (v > best) { best = v; bi = c; } }
  conf[i]  = (best < CONF_T) ? 0.f : best;
  label[i] = bi;

  const float* bx = boxes + (size_t)i * 5;
  float X = bx[0] - 320.f;           // constant shift: dx,dy unchanged, magnitudes halved
  float Y = bx[1] - 320.f;
  float w = bx[2], h = bx[3], r = bx[4];
  float a0 = w * w * (1.f / 12.f);
  float b0 = h * h * (1.f / 12.f);
  float cs = cosf(r), sn = sinf(r);
  float A_ = a0 * cs * cs + b0 * sn * sn;
  float B_ = a0 * sn * sn + b0 * cs * cs;
  float C_ = (a0 - b0) * cs * sn;
  float det = A_ * B_ - C_ * C_;
  sdv[i] = sqrtf(fmaxf(det, 0.f));
  float X2 = X * X, Y2 = Y * Y, XY = X * Y;

  float* Fi = F + (size_t)i * FSTRIDE;
  float* Gi = G + (size_t)i * FSTRIDE;
  // t1 numerator: (A_i+A_j)(Y_i-Y_j)^2 + (B_i+B_j)(X_i-X_j)^2
  Fi[0] = A_ * Y2 + B_ * X2;  Gi[0] = 1.f;
  Fi[1] = 1.f;                Gi[1] = A_ * Y2 + B_ * X2;
  Fi[2] = A_;                 Gi[2] = Y2;
  Fi[3] = Y2;                 Gi[3] = A_;
  Fi[4] = -2.f * A_ * Y;      Gi[4] = Y;
  Fi[5] = Y;                  Gi[5] = -2.f * A_ * Y;
  Fi[6] = B_;                 Gi[6] = X2;
  Fi[7] = X2;                 Gi[7] = B_;
  Fi[8] = -2.f * B_ * X;      Gi[8] = X;
  Fi[9] = X;                  Gi[9] = -2.f * B_ * X;
  Fi[10] = 0.f; Gi[10] = 0.f; Fi[11] = 0.f; Gi[11] = 0.f;
  // t2 numerator: -(C_i+C_j)(X_i-X_j)(Y_i-Y_j)
  Fi[12] = -C_ * XY;          Gi[12] = 1.f;
  Fi[13] =  C_ * X;           Gi[13] = Y;
  Fi[14] =  C_ * Y;           Gi[14] = X;
  Fi[15] = -C_;               Gi[15] = XY;
  Fi[16] = -XY;               Gi[16] = C_;
  Fi[17] =  X;                Gi[17] = C_ * Y;
  Fi[18] =  Y;                Gi[18] = C_ * X;
  Fi[19] = 1.f;               Gi[19] = -C_ * XY;
  // denom core: (A_i+A_j)(B_i+B_j) - (C_i+C_j)^2
  Fi[20] = det;               Gi[20] = 1.f;
  Fi[21] = 1.f;               Gi[21] = det;
  Fi[22] = A_;                Gi[22] = B_;
  Fi[23] = B_;                Gi[23] = A_;
  Fi[24] = -2.f * C_;         Gi[24] = C_;
  Fi[25] = 0.f; Gi[25] = 0.f; Fi[26] = 0.f; Gi[26] = 0.f; Fi[27] = 0.f; Gi[27] = 0.f;
}

// One wave owns a 16-column stripe; loops over 16-row tiles of the strict
// upper triangle. PASS 0: cmax[j] = max_{i<j} iou. PASS 1: dmax -> keep.
template <int PASS>
__global__ void k_pass(const float* __restrict__ F, const float* __restrict__ G,
                       const float* __restrict__ conf, float* __restrict__ cmax,
                       int* __restrict__ keep, const float* __restrict__ sdv) {
  __shared__ float lds_sd[NN];
  __shared__ float lds_cmax[NN];
  const int tid  = threadIdx.x;
  const int wave = tid >> 5;
  const int lane = tid & 31;
  const int stripe = blockIdx.x * 8 + wave;     // 8 waves per block, same batch
  const int b  = stripe >> 7;                   // 128 stripes per batch
  const int j0 = (stripe & 127) << 4;

  for (int t = tid; t < NN; t += blockDim.x) {
    lds_sd[t] = sdv[b * NN + t];
    if (PASS == 1) lds_cmax[t] = cmax[b * NN + t];
  }
  __syncthreads();

  const int half = lane >> 4;   // 0: feats k,k+1 ; 1: feats k+2,k+3 (A & B VGPR layout)
  const int l15  = lane & 15;

  // B matrix (column features) is loop-invariant: keep all 7 K-chunks in VGPRs.
  const size_t colbase = ((size_t)(b * NN + j0 + l15)) * FSTRIDE + 2 * half;
  v2f Bk[7];
#pragma unroll
  for (int k = 0; k < 7; ++k) Bk[k] = *(const v2f*)(G + colbase + 4 * k);
  const int gj = j0 + l15;
  const float sd_col = lds_sd[gj];

  float acc = 0.f;
  for (int m0 = 0; m0 <= j0; m0 += 16) {
    const size_t rowbase = ((size_t)(b * NN + m0 + l15)) * FSTRIDE + 2 * half;
    // keep the next A panel in flight (global_prefetch_b8)
    __builtin_prefetch(F + rowbase + 16 * FSTRIDE, 0, 1);

    v8f cT1 = {}; v8f cT2 = {}; v8f cD = {};
    v2f a;
    a = *(const v2f*)(F + rowbase + 0);  cT1 = wmma4(a, Bk[0], cT1);
    a = *(const v2f*)(F + rowbase + 4);  cT1 = wmma4(a, Bk[1], cT1);
    a = *(const v2f*)(F + rowbase + 8);  cT1 = wmma4(a, Bk[2], cT1);
    a = *(const v2f*)(F + rowbase + 12); cT2 = wmma4(a, Bk[3], cT2);
    a = *(const v2f*)(F + rowbase + 16); cT2 = wmma4(a, Bk[4], cT2);
    a = *(const v2f*)(F + rowbase + 20); cD  = wmma4(a, Bk[5], cD);
    a = *(const v2f*)(F + rowbase + 24); cD  = wmma4(a, Bk[6], cD);

    // Epilogue: C/D layout -> lane holds column gj, rows m0 + v + 8*half.
#pragma unroll
    for (int v = 0; v < 8; ++v) {
      const int gi = m0 + v + 8 * half;
      const float denomc = cD[v];
      const float rden = __builtin_amdgcn_rcpf(denomc + EPSF);
      const float t12  = (0.25f * cT1[v] + 0.5f * cT2[v]) * rden;
      const float sd_i = lds_sd[gi];
      const float arg  = denomc * __builtin_amdgcn_rcpf(4.f * sd_i * sd_col + EPSF) + EPSF;
      const float t3   = 0.5f * fast_ln(arg);
      const float bd   = fminf(fmaxf(t12 + t3, EPSF), 100.f);
      const float hd   = __builtin_amdgcn_sqrtf(1.f - fast_exp(-bd) + EPSF);
      const float iou  = 1.f - hd;
      const bool valid = gi < gj;   // strict upper triangle
      if (PASS == 0) {
        acc = fmaxf(acc, valid ? iou : 0.f);
      } else {
        const float cmi = lds_cmax[gi];
        acc = fmaxf(acc, valid ? (iou * iou - cmi * cmi) : 0.f);
      }
    }
  }

  // combine the two M-halves (lane l <-> l+16 hold same column)
  int other = __builtin_amdgcn_ds_swizzle(__float_as_int(acc), 0x401F); // SWAPX16
  acc = fmaxf(acc, __int_as_float(other));

  if (lane < 16) {
    const int j = j0 + lane;
    if (PASS == 0) {
      cmax[b * NN + j] = acc;
    } else {
      const float decay = fast_exp(-fmaxf(acc, 0.f) * (1.f / SIGMA));
      keep[b * NN + j] = (conf[b * NN + j] * decay > IOU_T) ? 1 : 0;
    }
  }
}

// Stable compaction: kept indices in ascending order fill the first slots;
// output was pre-zeroed so non-kept slots stay zero (matches masked rows).
__global__ void k_compact(const float* __restrict__ boxes, const float* __restrict__ conf,
                          const int* __restrict__ label, const int* __restrict__ keep,
                          float* __restrict__ out) {
  __shared__ int ssum[1024];
  const int b = blockIdx.x;
  const int t = threadIdx.x;
  const int i0 = 2 * t, i1 = 2 * t + 1;
  const int k0 = keep[b * NN + i0];
  const int k1 = keep[b * NN + i1];
  const int s  = k0 + k1;
  ssum[t] = s;
  __syncthreads();
  for (int off = 1; off < 1024; off <<= 1) {
    int v = (t >= off) ? ssum[t - off] : 0;
    __syncthreads();
    ssum[t] += v;
    __syncthreads();
  }
  const int excl = ssum[t] - s;

  auto write_row = [&](int idx, int pos) {
    const float* bx = boxes + ((size_t)b * NN + idx) * 5;
    float* o = out + ((size_t)b * MAX_PRED + pos) * 8;
    o[0] = (float)b;
    o[1] = bx[0]; o[2] = bx[1]; o[3] = bx[2]; o[4] = bx[3]; o[5] = bx[4];
    o[6] = conf[b * NN + idx];
    o[7] = (float)label[b * NN + idx];
  };
  if (k0 && excl < MAX_PRED)          write_row(i0, excl);
  if (k1 && (excl + k0) < MAX_PRED)   write_row(i1, excl + k0);
}

extern "C" void kernel_launch(void* const* d_in, const int* in_sizes, int n_in,
                              void* d_out, int out_size, void* d_ws, size_t ws_size,
                              hipStream_t stream) {
  (void)in_sizes; (void)n_in; (void)out_size; (void)ws_size;
  const float* boxes  = (const float*)d_in[0];   // (B,N,5) f32
  const float* scores = (const float*)d_in[1];   // (B,N,C) f32
  float* out = (float*)d_out;                    // (B*300, 8) f32

  char* ws = (char*)d_ws;
  const size_t featBytes = (size_t)BB * NN * FSTRIDE * sizeof(float); // 2 MB each
  const size_t vecBytes  = (size_t)BB * NN * sizeof(float);           // 64 KB each
  float* F    = (float*)(ws);
  float* G    = (float*)(ws + featBytes);
  float* conf = (float*)(ws + 2 * featBytes);
  float* cmax = (float*)(ws + 2 * featBytes + vecBytes);
  float* sdv  = (float*)(ws + 2 * featBytes + 2 * vecBytes);
  int*   lbl  = (int*)  (ws + 2 * featBytes + 3 * vecBytes);
  int*   keep = (int*)  (ws + 2 * featBytes + 4 * vecBytes);

  const int nOut = BB * MAX_PRED * 8;
  k_zero<<<(nOut + 255) / 256, 256, 0, stream>>>(out, nOut);
  k_prep<<<(BB * NN + 255) / 256, 256, 0, stream>>>(boxes, scores, F, G, conf, lbl, sdv);
  const int passBlocks = BB * (NN / 16) / 8;   // 128 blocks of 8 waves
  k_pass<0><<<passBlocks, 256, 0, stream>>>(F, G, conf, cmax, keep, sdv);
  k_pass<1><<<passBlocks, 256, 0, stream>>>(F, G, conf, cmax, keep, sdv);
  k_compact<<<BB, 1024, 0, stream>>>(boxes, conf, lbl, keep, out);
}